// FastACV_Plus_Original_34437047779557
// MI455X (gfx1250) — compile-verified
//
#include <hip/hip_runtime.h>

typedef __attribute__((ext_vector_type(2))) float v2f;
typedef __attribute__((ext_vector_type(8))) float v8f;

#define B_   2
#define C_   16
#define D_   48
#define K_   24
#define H4_  128
#define W4_  240
#define HW4  (H4_ * W4_)        /* 30720  */
#define NP4  (B_ * HW4)         /* 61440  */
#define H_   512
#define W_   960
#define HW1  (H_ * W_)          /* 491520 */
#define NP1  (B_ * HW1)         /* 983040 */

/* d_out layout (reference tuple order, flat):
   [0]            pred_att_x4  (NP4)
   [NP4]          pred_att_x1  (NP1)
   [NP4+NP1]      pred_x4      (NP4)
   [NP4+NP1+NP4]  pred_x1      (NP1) */
#define OUT_ATT4 0
#define OUT_ATT1 (NP4)
#define OUT_X4   (NP4 + NP1)
#define OUT_X1   (NP4 + NP1 + NP4)

// ---------------------------------------------------------------------------
// Kernel A: fp32 WMMA channel reduction.
//   L[p] = sum_c w_red[c]      * feat_l[c][p]
//   R[p] = sum_c w_red[C_+c]   * feat_r[c][p]
// One wave handles a tile of 16 pixels: pixels ride the A-matrix M dimension,
// channels ride K (4 per WMMA, 4 steps = 16 channels), w_red is broadcast
// across the B-matrix N columns. Result taken from the N=0 column of D.
// A-layout (ISA 7.12.2, 32-bit A 16x4): row M = lane&15, K = vgpr + 2*(lane>>4).
// B rows follow the same vgpr/lane-half striping, so b[vgpr] = w[k] with the
// identical k formula; since every N column is the same weight, lane's N is
// irrelevant.
// ---------------------------------------------------------------------------
__global__ __launch_bounds__(256) void reduce_feats_wmma(
    const float* __restrict__ feat_l, const float* __restrict__ feat_r,
    const float* __restrict__ w_red,
    float* __restrict__ Lw, float* __restrict__ Rw)
{
    const int lane = threadIdx.x & 31;
    const int wave = threadIdx.x >> 5;
    const int tile = blockIdx.x * 8 + wave;      // 16 pixels per wave-tile
    const int base = tile * 16;                  // NP4 % 16 == 0
    const int m    = lane & 15;                  // A-matrix row (pixel in tile)
    const int h    = lane >> 4;                  // lane half selects K group
    const int p    = base + m;
    const int b    = p / HW4;
    const int rem  = p - b * HW4;

    const float* fl = feat_l + (size_t)b * (C_ * HW4) + rem;
    const float* fr = feat_r + (size_t)b * (C_ * HW4) + rem;

    v8f accL = {};
    v8f accR = {};
#pragma unroll
    for (int s = 0; s < 4; ++s) {
        const int c0 = s * 4 + 2 * h;            // k = vgpr + 2*h  ->  channels c0, c0+1
        v2f aL, aR, bL, bR;
        aL.x = fl[(size_t)c0 * HW4];       aL.y = fl[(size_t)(c0 + 1) * HW4];
        aR.x = fr[(size_t)c0 * HW4];       aR.y = fr[(size_t)(c0 + 1) * HW4];
        bL.x = w_red[c0];                  bL.y = w_red[c0 + 1];
        bR.x = w_red[C_ + c0];             bR.y = w_red[C_ + c0 + 1];
        accL = __builtin_amdgcn_wmma_f32_16x16x4_f32(false, aL, false, bL,
                                                     (short)0, accL, false, false);
        accR = __builtin_amdgcn_wmma_f32_16x16x4_f32(false, aR, false, bR,
                                                     (short)0, accR, false, false);
    }

    // D layout: vgpr v, lanes 0-15 -> (M=v, N=lane); lanes 16-31 -> (M=v+8, N=lane-16).
    // Take the N==0 column: lane 0 holds rows 0..7, lane 16 holds rows 8..15.
    if (m == 0) {
#pragma unroll
        for (int v = 0; v < 8; ++v) {
            Lw[base + v + 8 * h] = accL[v];
            Rw[base + v + 8 * h] = accR[v];
        }
    }
}

// ---------------------------------------------------------------------------
// Kernel B: per low-res pixel. 48 att values live in registers (all loops
// fully unrolled; top-24 kept as a bitmask so nothing is dynamically indexed).
// ---------------------------------------------------------------------------
__global__ __launch_bounds__(256) void lowres_kernel(
    const float* __restrict__ att,
    const float* __restrict__ Lw, const float* __restrict__ Rw,
    float* __restrict__ out)
{
    const int p = blockIdx.x * blockDim.x + threadIdx.x;
    if (p >= NP4) return;
    const int b   = p / HW4;
    const int rem = p - b * HW4;
    const int x   = rem % W4_;

    const float* ap = att + (size_t)b * (D_ * HW4) + rem;

    float vals[D_];
    float M48 = -__builtin_inff();
#pragma unroll
    for (int d = 0; d < D_; ++d) {
        vals[d] = ap[(size_t)d * HW4];
        M48 = fmaxf(M48, vals[d]);
    }
    float Z48 = 0.f;
#pragma unroll
    for (int d = 0; d < D_; ++d) Z48 += __expf(vals[d] - M48);

    // top-24 of 48 by value (softmax is monotone -> rank on raw values),
    // jax tie-break: lower index wins on equal values.
    unsigned long long keep = 0ull;
#pragma unroll
    for (int i = 0; i < D_; ++i) {
        int rank = 0;
#pragma unroll
        for (int j = 0; j < D_; ++j)
            rank += (vals[j] > vals[i]) || ((vals[j] == vals[i]) && (j < i));
        if (rank < K_) keep |= (1ull << i);
    }

    // ---- attention-only regression: softmax over the 24 kept raw values ----
    float mk = -__builtin_inff();
#pragma unroll
    for (int i = 0; i < D_; ++i)
        if (keep & (1ull << i)) mk = fmaxf(mk, vals[i]);
    float se = 0.f, sed = 0.f;
#pragma unroll
    for (int i = 0; i < D_; ++i)
        if (keep & (1ull << i)) {
            const float e = __expf(vals[i] - mk);
            se += e;
            sed += e * (float)i;
        }
    const float pred_att = sed / se;

    // ---- sparse cost over kept disparities + streaming top-2 ----
    // cost(d) = att_topk(d) * (L[p] + valid * R[p - d])
    const float Lp = Lw[p];
    float c1 = -__builtin_inff(), c2 = -__builtin_inff();
    float d1 = 0.f, d2 = 0.f;
#pragma unroll
    for (int i = 0; i < D_; ++i)
        if (keep & (1ull << i)) {
            const float attp = __expf(vals[i] - M48) / Z48;
            const float r    = (x - i >= 0) ? Rw[p - i] : 0.f;
            const float cost = attp * (Lp + r);
            if (cost > c1)      { c2 = c1; d2 = d1; c1 = cost; d1 = (float)i; }
            else if (cost > c2) { c2 = cost; d2 = (float)i; }
        }
    const float p1   = 1.f / (1.f + __expf(c2 - c1));   // softmax of 2
    const float pred = p1 * d1 + (1.f - p1) * d2;

    out[OUT_ATT4 + p] = pred_att;
    out[OUT_X4   + p] = pred;
}

// ---------------------------------------------------------------------------
// Kernel C: superpixel-guided 4x upsample of both low-res predictions.
// neigh_j(y,x) = disp_low[y/4 + dy, x/4 + dx], j = (dy+1)*3 + (dx+1), zero pad.
// ---------------------------------------------------------------------------
__global__ __launch_bounds__(256) void upsample_kernel(
    const float* __restrict__ spx,
    const float* __restrict__ low_att,   // = out + OUT_ATT4
    const float* __restrict__ low_x,     // = out + OUT_X4
    float* __restrict__ out)
{
    const int p = blockIdx.x * blockDim.x + threadIdx.x;
    if (p >= NP1) return;
    const int b   = p / HW1;
    const int rem = p - b * HW1;
    const int y   = rem / W_;
    const int x   = rem - y * W_;

    const float* sp = spx + (size_t)b * (9 * HW1) + rem;
    float e[9];
    float m = -__builtin_inff();
#pragma unroll
    for (int j = 0; j < 9; ++j) {
        e[j] = sp[(size_t)j * HW1];
        m = fmaxf(m, e[j]);
    }
    float Z = 0.f;
#pragma unroll
    for (int j = 0; j < 9; ++j) { e[j] = __expf(e[j] - m); Z += e[j]; }

    const int ly = y >> 2, lx = x >> 2;
    const float* A = low_att + (size_t)b * HW4;
    const float* X = low_x   + (size_t)b * HW4;

    float sa = 0.f, sx = 0.f;
#pragma unroll
    for (int dy = -1; dy <= 1; ++dy)
#pragma unroll
        for (int dx = -1; dx <= 1; ++dx) {
            const int j  = (dy + 1) * 3 + (dx + 1);
            const int yy = ly + dy, xx = lx + dx;
            const bool in = (yy >= 0) & (yy < H4_) & (xx >= 0) & (xx < W4_);
            const float va = in ? A[yy * W4_ + xx] : 0.f;
            const float vx = in ? X[yy * W4_ + xx] : 0.f;
            sa += e[j] * va;
            sx += e[j] * vx;
        }

    const float inv = 4.f / Z;
    out[OUT_ATT1 + p] = sa * inv;
    out[OUT_X1   + p] = sx * inv;
}

// ---------------------------------------------------------------------------
extern "C" void kernel_launch(void* const* d_in, const int* in_sizes, int n_in,
                              void* d_out, int out_size, void* d_ws, size_t ws_size,
                              hipStream_t stream) {
    (void)in_sizes; (void)n_in; (void)out_size; (void)ws_size;
    const float* att    = (const float*)d_in[0];   // [B,1,48,128,240]
    const float* feat_l = (const float*)d_in[1];   // [B,16,128,240]
    const float* feat_r = (const float*)d_in[2];   // [B,16,128,240]
    const float* w_red  = (const float*)d_in[3];   // [32]
    const float* spx    = (const float*)d_in[4];   // [B,9,512,960]
    float*       out    = (float*)d_out;

    float* Lw = (float*)d_ws;          // NP4 floats
    float* Rw = Lw + NP4;              // NP4 floats  (480 KB total)

    // A: WMMA channel reduction -> L, R  (3840 wave-tiles, 8 waves/block)
    reduce_feats_wmma<<<NP4 / (16 * 8), 256, 0, stream>>>(feat_l, feat_r, w_red, Lw, Rw);

    // B: per low-res pixel pipeline
    lowres_kernel<<<(NP4 + 255) / 256, 256, 0, stream>>>(att, Lw, Rw, out);

    // C: superpixel 4x upsample (reads low-res predictions written by B)
    upsample_kernel<<<(NP1 + 255) / 256, 256, 0, stream>>>(
        spx, out + OUT_ATT4, out + OUT_X4, out);
}